// SGC_91250875171026
// MI455X (gfx1250) — compile-verified
//
#include <hip/hip_runtime.h>
#include <hip/hip_bf16.h>

typedef __attribute__((ext_vector_type(2))) float v2f;
typedef __attribute__((ext_vector_type(8))) float v8f;

#define DFEAT 128

// ---------------- init: deg=1 (self loop), h1=0, out=0 ----------------
__global__ void sgc_init_kernel(float* __restrict__ deg, float* __restrict__ h1,
                                float* __restrict__ out, int n) {
    int i = blockIdx.x * blockDim.x + threadIdx.x;
    if (i < n) { deg[i] = 1.0f; h1[i] = 0.0f; out[i] = 0.0f; }
}

// ---------------- z = x @ W^T via V_WMMA_F32_16X16X4_F32 ----------------
// One wave handles a 16-row tile of x. A = x[16 x 4] chunk, B = W chunk in
// column N=0 only. After 32 WMMAs over K=128, column 0 of D holds the dots:
// lane 0 -> rows M=0..7 in c[0..7], lane 16 -> rows M=8..15.
__global__ void sgc_zdot_wmma(const float* __restrict__ x,
                              const float* __restrict__ W,
                              float* __restrict__ z, int n) {
    int wave = (int)((blockIdx.x * (unsigned)blockDim.x + threadIdx.x) >> 5);
    int lane = (int)(threadIdx.x & 31u);
    int row0 = wave * 16;
    if (row0 >= n) return;                 // wave-uniform: EXEC stays all-1s
    int m  = lane & 15;                    // row within tile (A) / col (B)
    int kh = (lane >> 4) << 1;             // half-lanes carry K=0,1 vs K=2,3
    int r  = row0 + m; if (r >= n) r = n - 1;   // clamp tail reads
    const float* xr = x + (size_t)r * DFEAT;

    v8f c = {};
    #pragma unroll 4
    for (int k = 0; k < DFEAT; k += 4) {
        v2f a, bm;
        a.x = xr[k + kh];
        a.y = xr[k + kh + 1];
        float w0 = W[k + kh];
        float w1 = W[k + kh + 1];
        bm.x = (m == 0) ? w0 : 0.0f;       // B nonzero only in column N=0
        bm.y = (m == 0) ? w1 : 0.0f;
        // (neg_a, A, neg_b, B, c_mod, C, reuse_a, reuse_b)
        c = __builtin_amdgcn_wmma_f32_16x16x4_f32(
                false, a, false, bm, (short)0, c, false, false);
    }
    if (m == 0) {
        int base = row0 + ((lane >> 4) << 3);
        #pragma unroll
        for (int v = 0; v < 8; ++v)
            if (base + v < n) z[base + v] = c[v];
    }
}

// ---------------- degree accumulation over explicit edges ----------------
__global__ void sgc_deg_kernel(const int* __restrict__ dst,
                               float* __restrict__ deg, int e) {
    int i = blockIdx.x * blockDim.x + threadIdx.x;
    if (i < e) atomicAdd(&deg[dst[i]], 1.0f);
}

// ---------------- dis = rsqrt(deg) in place (deg >= 1 always) ------------
__global__ void sgc_rsqrt_kernel(float* __restrict__ deg, int n) {
    int i = blockIdx.x * blockDim.x + threadIdx.x;
    if (i < n) deg[i] = rsqrtf(deg[i]);
}

// ---------------- one propagation hop: hout[d] += dis[s]*dis[d]*hin[s] ----
// Covers E explicit edges plus N self-loops (i >= e maps to node i-e).
__global__ void sgc_hop_kernel(const int* __restrict__ src,
                               const int* __restrict__ dst,
                               const float* __restrict__ dis,
                               const float* __restrict__ hin,
                               float* __restrict__ hout, int e, int n) {
    int i = blockIdx.x * blockDim.x + threadIdx.x;
    if (i < e) {
        int s = src[i];
        int d = dst[i];
        atomicAdd(&hout[d], dis[s] * dis[d] * hin[s]);
    } else if (i < e + n) {
        int s = i - e;
        float ds = dis[s];
        atomicAdd(&hout[s], ds * ds * hin[s]);
    }
}

// ---------------- out += b ----------------
__global__ void sgc_bias_kernel(float* __restrict__ out,
                                const float* __restrict__ b, int n) {
    int i = blockIdx.x * blockDim.x + threadIdx.x;
    if (i < n) out[i] += b[0];
}

extern "C" void kernel_launch(void* const* d_in, const int* in_sizes, int n_in,
                              void* d_out, int out_size, void* d_ws, size_t ws_size,
                              hipStream_t stream) {
    (void)n_in; (void)out_size; (void)ws_size;
    const float* x  = (const float*)d_in[0];   // [N, 128]
    const int*   ei = (const int*)d_in[1];     // [2, E]
    const float* W  = (const float*)d_in[2];   // [1, 128]
    const float* b  = (const float*)d_in[3];   // [1]
    float* out = (float*)d_out;                // [N]

    int n = in_sizes[0] / DFEAT;
    int e = in_sizes[1] / 2;
    const int* src = ei;
    const int* dst = ei + e;

    float* deg = (float*)d_ws;   // n floats (becomes dis after rsqrt)
    float* z   = deg + n;        // n floats
    float* h1  = z + n;          // n floats

    const int tb = 256;
    int gb_n  = (n + tb - 1) / tb;
    int gb_e  = (e + tb - 1) / tb;
    int gb_en = (e + n + tb - 1) / tb;

    sgc_init_kernel<<<gb_n, tb, 0, stream>>>(deg, h1, out, n);

    int tiles = (n + 15) / 16;                 // one wave (32 lanes) per tile
    int waves_per_block = tb / 32;
    int gb_wmma = (tiles + waves_per_block - 1) / waves_per_block;
    sgc_zdot_wmma<<<gb_wmma, tb, 0, stream>>>(x, W, z, n);

    sgc_deg_kernel<<<gb_e, tb, 0, stream>>>(dst, deg, e);
    sgc_rsqrt_kernel<<<gb_n, tb, 0, stream>>>(deg, n);
    sgc_hop_kernel<<<gb_en, tb, 0, stream>>>(src, dst, deg, z,  h1,  e, n);
    sgc_hop_kernel<<<gb_en, tb, 0, stream>>>(src, dst, deg, h1, out, e, n);
    sgc_bias_kernel<<<gb_n, tb, 0, stream>>>(out, b, n);
}